// GIN_8718783611640
// MI455X (gfx1250) — compile-verified
//
#include <hip/hip_runtime.h>
#include <hip/hip_bf16.h>
#include <cstdint>

#define DIM 128
#define N_NODES_C 50000
#define N_EDGES_C 800000
#define N_GRAPHS_C 128
#define N_CLASSES_C 10
#define LDA_PAD 132      // LDS stride (floats) for A tile: conflict-free frag reads
#define WP_STRIDE_F 288  // floats per K-pair row of W (144 float2): 288 dwords = 32 mod 64 banks

typedef float v2f __attribute__((ext_vector_type(2)));
typedef float v8f __attribute__((ext_vector_type(8)));

// ---------------- elementwise helpers ----------------

__global__ void gin_copy4_kernel(const float* __restrict__ s, float* __restrict__ d, int n4) {
    int i = blockIdx.x * blockDim.x + threadIdx.x;
    if (i < n4) {
        const float4 v = ((const float4*)s)[i];
        ((float4*)d)[i] = v;
    }
}

__global__ void gin_zero_kernel(float* __restrict__ d, int n) {
    int i = blockIdx.x * blockDim.x + threadIdx.x;
    if (i < n) d[i] = 0.0f;
}

// ---------------- edge scatter-add (message passing) ----------------
// one wave (32 lanes) per edge; each lane moves 4 contiguous floats
__global__ __launch_bounds__(256) void gin_scatter_kernel(
    const float* __restrict__ feat, const int* __restrict__ src_idx,
    const int* __restrict__ dst_idx, float* __restrict__ agg, int nedges)
{
    int gid  = blockIdx.x * blockDim.x + threadIdx.x;
    int e    = gid >> 5;
    int lane = gid & 31;
    if (e >= nedges) return;
    int s = src_idx[e];
    int t = dst_idx[e];
    const float4 v = *(const float4*)(feat + (size_t)s * DIM + lane * 4);
    float* o = agg + (size_t)t * DIM + lane * 4;
    atomicAdd(o + 0, v.x);
    atomicAdd(o + 1, v.y);
    atomicAdd(o + 2, v.z);
    atomicAdd(o + 3, v.w);
}

// ---------------- WMMA fp32 GEMM: C = act(A[nrows x 128] @ W[128 x 128] + b) ----------------
// block = 256 threads = 8 waves; block computes 128 output rows.
// Wave w owns rows [w*16, w*16+16) and all 8 column tiles (16 cols each).
// V_WMMA_F32_16X16X4_F32: A frag 16x4, B frag 4x16, C/D 16x16, 32 K-steps.
// A tile staged via GLOBAL_LOAD_ASYNC_TO_LDS_B128 (ASYNCcnt path, per-lane padded dests);
// W staged K-pair-interleaved so each B fragment is one aligned ds_load_b64.
__global__ __launch_bounds__(256) void gin_mlp_gemm_kernel(
    const float* __restrict__ A, const float* __restrict__ W,
    const float* __restrict__ bias, float* __restrict__ C,
    int nrows, int do_relu)
{
    extern __shared__ float smem[];
    float* Alds = smem;                       // 128 * LDA_PAD floats
    float* Wp   = smem + 128 * LDA_PAD;       // 64 pair-rows * WP_STRIDE_F floats
    float* blds = Wp + 64 * WP_STRIDE_F;      // 128 floats

    const int t    = threadIdx.x;
    const int row0 = blockIdx.x * 128;

    // ---- stage A tile (128 rows x 32 cols) via async global->LDS, padded stride ----
    // 1024 16-byte chunks, 4 per thread; OOB tail rows clamped (their outputs are store-masked)
    for (int i = t; i < 128 * 8; i += 256) {
        int r  = i >> 3;
        int c4 = (i & 7) * 4;
        int gr = row0 + r;
        if (gr >= nrows) gr = nrows - 1;
        const float* gsrc = A + (size_t)gr * DIM + c4;
        unsigned lds_off  = (unsigned)(uintptr_t)&Alds[r * LDA_PAD + c4];
        asm volatile("global_load_async_to_lds_b128 %0, %1, off"
                     :: "v"(lds_off), "v"(gsrc)
                     : "memory");
    }

    // ---- stage W as K-pair interleave: Wp[p][n] = (W[2p][n], W[2p+1][n]) ----
    for (int i = t; i < 64 * 32; i += 256) {
        int p = i >> 5;            // pair row 0..63
        int n = (i & 31) * 4;      // column base
        const float4 w0 = *(const float4*)(W + (size_t)(2 * p) * DIM + n);
        const float4 w1 = *(const float4*)(W + (size_t)(2 * p + 1) * DIM + n);
        float* dst = &Wp[p * WP_STRIDE_F + n * 2];
        *(float4*)(dst + 0) = make_float4(w0.x, w1.x, w0.y, w1.y);
        *(float4*)(dst + 4) = make_float4(w0.z, w1.z, w0.w, w1.w);
    }
    if (t < 128) blds[t] = bias[t];

    asm volatile("s_wait_asynccnt 0" ::: "memory");
    __syncthreads();

    const int wave = t >> 5;        // 0..7
    const int lane = t & 31;
    const int half = lane >> 4;     // 0 or 1
    const int l16  = lane & 15;
    const int m0   = wave * 16;     // row strip within block

    v8f acc[8];
#pragma unroll
    for (int j = 0; j < 8; ++j) acc[j] = (v8f){0.f, 0.f, 0.f, 0.f, 0.f, 0.f, 0.f, 0.f};

    const float* arow = &Alds[(m0 + l16) * LDA_PAD];
    for (int k0 = 0; k0 < DIM; k0 += 4) {
        // A frag (ISA 7.12.2): lane half 0 -> K={k0,k0+1}, half 1 -> K={k0+2,k0+3}
        v2f a = *(const v2f*)(arow + k0 + 2 * half);
        // B frag: pair row p = k0/2 + half holds (W[k][n], W[k+1][n]) contiguously
        const float* wrow = &Wp[((k0 >> 1) + half) * WP_STRIDE_F];
#pragma unroll
        for (int j = 0; j < 8; ++j) {
            v2f b = *(const v2f*)&wrow[(j * 16 + l16) * 2];
            acc[j] = __builtin_amdgcn_wmma_f32_16x16x4_f32(
                false, a, false, b, (short)0, acc[j], false, false);
        }
    }

    // C/D layout: VGPR r holds M = r + 8*half, N = l16 (per 16-col tile)
#pragma unroll
    for (int j = 0; j < 8; ++j) {
        int col  = j * 16 + l16;
        float bb = blds[col];
#pragma unroll
        for (int r = 0; r < 8; ++r) {
            int grow = row0 + m0 + r + 8 * half;
            if (grow < nrows) {
                float v = acc[j][r] + bb;
                if (do_relu) v = fmaxf(v, 0.f);
                C[(size_t)grow * DIM + col] = v;
            }
        }
    }
}

// ---------------- BatchNorm statistics (sum, sumsq per column) ----------------
__global__ __launch_bounds__(256) void gin_bn_stats_kernel(
    const float* __restrict__ h, float* __restrict__ sums, float* __restrict__ sqs, int nrows)
{
    const int c  = threadIdx.x & 127;
    const int rh = threadIdx.x >> 7;   // 0/1
    float s = 0.f, q = 0.f;
    for (int r = blockIdx.x * 2 + rh; r < nrows; r += gridDim.x * 2) {
        float v = h[(size_t)r * DIM + c];
        s += v;
        q += v * v;
    }
    __shared__ float ls[256], lq[256];
    ls[threadIdx.x] = s;
    lq[threadIdx.x] = q;
    __syncthreads();
    if (threadIdx.x < 128) {
        atomicAdd(&sums[c], ls[c] + ls[c + 128]);
        atomicAdd(&sqs[c],  lq[c] + lq[c + 128]);
    }
}

// ---------------- BatchNorm apply + ReLU ----------------
__global__ void gin_bn_apply_kernel(
    const float* __restrict__ h, float* __restrict__ o,
    const float* __restrict__ sums, const float* __restrict__ sqs,
    const float* __restrict__ gamma, const float* __restrict__ beta, int nrows)
{
    int i = blockIdx.x * blockDim.x + threadIdx.x;
    int n = nrows * DIM;
    if (i >= n) return;
    int c = i & (DIM - 1);
    float invn = 1.0f / (float)nrows;
    float mu   = sums[c] * invn;
    float var  = fmaxf(sqs[c] * invn - mu * mu, 0.f);
    float v = (h[i] - mu) * rsqrtf(var + 1e-5f) * gamma[c] + beta[c];
    o[i] = fmaxf(v, 0.f);
}

// ---------------- global add pool: one wave per node ----------------
__global__ __launch_bounds__(256) void gin_pool_kernel(
    const float* __restrict__ h, const int* __restrict__ batch,
    float* __restrict__ pooled, int nrows)
{
    int gid  = blockIdx.x * blockDim.x + threadIdx.x;
    int node = gid >> 5;
    int lane = gid & 31;
    if (node >= nrows) return;
    int g = batch[node];
    const float4 v = *(const float4*)(h + (size_t)node * DIM + lane * 4);
    float* o = pooled + (size_t)g * DIM + lane * 4;
    atomicAdd(o + 0, v.x);
    atomicAdd(o + 1, v.y);
    atomicAdd(o + 2, v.z);
    atomicAdd(o + 3, v.w);
}

// ---------------- final tiny GEMM: out[128 x 10] = pooled @ Wo + bo ----------------
__global__ void gin_out_gemm_kernel(
    const float* __restrict__ pooled, const float* __restrict__ Wo,
    const float* __restrict__ bo, float* __restrict__ out)
{
    int t = blockIdx.x * blockDim.x + threadIdx.x;
    if (t >= N_GRAPHS_C * N_CLASSES_C) return;
    int g = t / N_CLASSES_C;
    int c = t % N_CLASSES_C;
    float s = bo[c];
#pragma unroll 8
    for (int k = 0; k < DIM; ++k) s += pooled[g * DIM + k] * Wo[k * N_CLASSES_C + c];
    out[t] = s;
}

extern "C" void kernel_launch(void* const* d_in, const int* in_sizes, int n_in,
                              void* d_out, int out_size, void* d_ws, size_t ws_size,
                              hipStream_t stream) {
    (void)in_sizes; (void)n_in; (void)out_size; (void)ws_size;

    const float* x   = (const float*)d_in[0];
    const int*   ei  = (const int*)d_in[1];        // [2, E]
    const int*   bat = (const int*)d_in[2];        // [N]
    const float* W1a = (const float*)d_in[3];
    const float* b1a = (const float*)d_in[4];
    const float* W1b = (const float*)d_in[5];
    const float* b1b = (const float*)d_in[6];
    const float* g1  = (const float*)d_in[7];
    const float* be1 = (const float*)d_in[8];
    const float* W2a = (const float*)d_in[9];
    const float* b2a = (const float*)d_in[10];
    const float* W2b = (const float*)d_in[11];
    const float* b2b = (const float*)d_in[12];
    const float* g2  = (const float*)d_in[13];
    const float* be2 = (const float*)d_in[14];
    const float* Wo  = (const float*)d_in[15];
    const float* bo  = (const float*)d_in[16];
    float* out = (float*)d_out;

    const int N = N_NODES_C, E = N_EDGES_C;
    const int* esrc = ei;
    const int* edst = ei + E;

    // workspace partition (floats)
    float* buf0   = (float*)d_ws;                    // [N, 128]
    float* buf1   = buf0 + (size_t)N * DIM;          // [N, 128]
    float* sums   = buf1 + (size_t)N * DIM;          // [128]
    float* sqs    = sums + DIM;                      // [128]
    float* pooled = sqs + DIM;                       // [128, 128]

    const int n4          = N * DIM / 4;
    const int copy_blocks = (n4 + 255) / 256;
    const int scat_blocks = (E * 32 + 255) / 256;
    const int gemm_blocks = (N + 127) / 128;
    const int elem_blocks = (N * DIM + 255) / 256;
    const int pool_blocks = (N * 32 + 255) / 256;
    const size_t gemm_lds = (size_t)(128 * LDA_PAD + 64 * WP_STRIDE_F + 128) * sizeof(float);

    // ---- layer 1: agg = x + scatter_add(x) ----
    gin_copy4_kernel<<<copy_blocks, 256, 0, stream>>>(x, buf0, n4);
    gin_scatter_kernel<<<scat_blocks, 256, 0, stream>>>(x, esrc, edst, buf0, E);
    // MLP1
    gin_mlp_gemm_kernel<<<gemm_blocks, 256, gemm_lds, stream>>>(buf0, W1a, b1a, buf1, N, 1);
    gin_mlp_gemm_kernel<<<gemm_blocks, 256, gemm_lds, stream>>>(buf1, W1b, b1b, buf0, N, 0);
    // BN1 + ReLU
    gin_zero_kernel<<<1, 256, 0, stream>>>(sums, 2 * DIM);
    gin_bn_stats_kernel<<<256, 256, 0, stream>>>(buf0, sums, sqs, N);
    gin_bn_apply_kernel<<<elem_blocks, 256, 0, stream>>>(buf0, buf1, sums, sqs, g1, be1, N);

    // ---- layer 2: agg = h + scatter_add(h) ----
    gin_copy4_kernel<<<copy_blocks, 256, 0, stream>>>(buf1, buf0, n4);
    gin_scatter_kernel<<<scat_blocks, 256, 0, stream>>>(buf1, esrc, edst, buf0, E);
    // MLP2
    gin_mlp_gemm_kernel<<<gemm_blocks, 256, gemm_lds, stream>>>(buf0, W2a, b2a, buf1, N, 1);
    gin_mlp_gemm_kernel<<<gemm_blocks, 256, gemm_lds, stream>>>(buf1, W2b, b2b, buf0, N, 0);
    // BN2 + ReLU
    gin_zero_kernel<<<1, 256, 0, stream>>>(sums, 2 * DIM);
    gin_bn_stats_kernel<<<256, 256, 0, stream>>>(buf0, sums, sqs, N);
    gin_bn_apply_kernel<<<elem_blocks, 256, 0, stream>>>(buf0, buf1, sums, sqs, g2, be2, N);

    // ---- pool + classifier ----
    gin_zero_kernel<<<(N_GRAPHS_C * DIM + 255) / 256, 256, 0, stream>>>(pooled, N_GRAPHS_C * DIM);
    gin_pool_kernel<<<pool_blocks, 256, 0, stream>>>(buf1, bat, pooled, N);
    gin_out_gemm_kernel<<<5, 256, 0, stream>>>(pooled, Wo, bo, out);
}